// MFPoolLayer_16363825397838
// MI455X (gfx1250) — compile-verified
//
#include <hip/hip_runtime.h>
#include <cfloat>

// Problem constants (from reference): N, M, C, B, K
#define NN 16384
#define MM 4096
#define CC 256
#define BB 8
#define KK 48

// Native clang vector so __builtin_nontemporal_store accepts it
// (HIP's float4 is a struct and is rejected by the builtin).
typedef float v4f __attribute__((ext_vector_type(4)));

// One block per output point m. 256 threads = 8 wave32 waves; wave w handles
// batch b = w. Each lane accumulates 8 channels as two 128-bit vectors, so one
// wave covers the full C=256 row with two contiguous 512B global_load_b128
// bursts.
//
// Compaction of the masked neighbor list is parallel: threads 0..47 load
// idx/mask with coalesced vector loads (hoisted so both issue before one
// wait) and claim output slots with an LDS atomicAdd, which the compiler
// wave-aggregates (mbcnt + single ds_add_rtn_u32). Order of compacted
// indices is irrelevant because max-reduction is commutative.
//
// Uold (128MB) is L2-resident on MI455X (192MB L2); the main loop is an
// L2-bandwidth-bound gather: widest VMEM loads straight to VGPRs, no LDS
// staging (gathered rows have no intra-block reuse; query points are random
// so there is no cross-m spatial locality either). Output stores are
// non-temporal: written once, never re-read — keep L2 for the gather stream.
__global__ __launch_bounds__(256)
void mfpool_max_kernel(const float* __restrict__ Uold,
                       const int* __restrict__ idx,
                       const unsigned char* __restrict__ mask,  // jnp bool = 1 byte
                       float* __restrict__ out)
{
    __shared__ int lidx[KK];
    __shared__ int lcnt;

    const int m    = blockIdx.x;
    const int tid  = threadIdx.x;
    const int lane = tid & 31;

    if (tid == 0) lcnt = 0;
    __syncthreads();

    // ---- parallel compaction of valid neighbor indices into LDS ----
    int e  = 0;
    int mk = 0;
    if (tid < KK) {
        e  = idx[m * KK + tid];          // coalesced dword load   (issued
        mk = (int)mask[m * KK + tid];    // coalesced byte load     together)
    }
    if (mk) {
        const int pos = atomicAdd(&lcnt, 1);   // wave-aggregated ds_add_rtn_u32
        lidx[pos] = e;
    }
    __syncthreads();

    const int cnt = lcnt;
    const int b   = tid >> 5;   // wave id == batch index

    // Base of this batch's [N, C] slice, viewed as 128-bit rows of 64 vectors.
    const v4f* __restrict__ Ub =
        (const v4f*)(Uold + (size_t)b * ((size_t)NN * CC));

    v4f acc0 = { -FLT_MAX, -FLT_MAX, -FLT_MAX, -FLT_MAX };
    v4f acc1 = acc0;

    // Branch-free gather/max over valid neighbors only.
    #pragma unroll 4
    for (int k = 0; k < cnt; ++k) {
        const int ee   = lidx[k];
        const int base = ee * (CC / 4);       // vec4 index; max ~1M, fits i32
        const v4f v0 = Ub[base + lane];       // bytes [0,   512) of the row
        const v4f v1 = Ub[base + 32 + lane];  // bytes [512,1024) of the row

        acc0.x = fmaxf(acc0.x, v0.x);
        acc0.y = fmaxf(acc0.y, v0.y);
        acc0.z = fmaxf(acc0.z, v0.z);
        acc0.w = fmaxf(acc0.w, v0.w);
        acc1.x = fmaxf(acc1.x, v1.x);
        acc1.y = fmaxf(acc1.y, v1.y);
        acc1.z = fmaxf(acc1.z, v1.z);
        acc1.w = fmaxf(acc1.w, v1.w);
    }

    // out[b, m, :] — two contiguous 512B non-temporal bursts per wave.
    v4f* __restrict__ Ob =
        (v4f*)(out + (size_t)b * ((size_t)MM * CC) + (size_t)m * CC);
    __builtin_nontemporal_store(acc0, Ob + lane);
    __builtin_nontemporal_store(acc1, Ob + lane + 32);
}

extern "C" void kernel_launch(void* const* d_in, const int* in_sizes, int n_in,
                              void* d_out, int out_size, void* d_ws, size_t ws_size,
                              hipStream_t stream) {
    // setup_inputs() order: Uold (f32 [B,N,C]), idx (i32 [M,K]), mask (bool [M,K])
    const float*         Uold = (const float*)d_in[0];
    const int*           idx  = (const int*)d_in[1];
    const unsigned char* mask = (const unsigned char*)d_in[2];
    float*               out  = (float*)d_out;

    mfpool_max_kernel<<<dim3(MM), dim3(256), 0, stream>>>(Uold, idx, mask, out);
}